// MVDRBeamformer_86560770883795
// MI455X (gfx1250) — compile-verified
//
#include <hip/hip_runtime.h>
#include <math.h>

typedef __attribute__((ext_vector_type(2))) float v2f;
typedef __attribute__((ext_vector_type(8))) float v8f;

#define T_FRAMES 512
#define F_FULL 513
#define N_MICS 8
#define N_ANG 36
#define I0 20
#define NF 428
#define NW 61
#define NSNAP 32
#define STEP8 8
#define DIAG_LOAD 0.001f
#define PI_D 3.14159265358979323846

struct cplx { float x, y; };
__device__ __forceinline__ cplx cadd(cplx a, cplx b){ return {a.x+b.x, a.y+b.y}; }
__device__ __forceinline__ cplx csub(cplx a, cplx b){ return {a.x-b.x, a.y-b.y}; }
__device__ __forceinline__ cplx cmul(cplx a, cplx b){ return {a.x*b.x - a.y*b.y, a.x*b.y + a.y*b.x}; }
__device__ __forceinline__ cplx cconj(cplx a){ return {a.x, -a.y}; }
__device__ __forceinline__ cplx cscale(cplx a, float s){ return {a.x*s, a.y*s}; }

// ---------------------------------------------------------------------------
// Kernel 1: steering vectors V[f][a][m] = exp(i * k(f,a) . p_m), interleaved re/im
// ---------------------------------------------------------------------------
__global__ void steer_kernel(const float* __restrict__ pos, float* __restrict__ V) {
  int idx = blockIdx.x * blockDim.x + threadIdx.x;
  if (idx >= NF * N_ANG) return;
  int f = idx / N_ANG, a = idx % N_ANG;
  float fhz = (float)(I0 + f) * (16000.0f / 1024.0f);
  float ang = (float)((double)a * (PI_D / 35.0));
  float ca = cosf(ang), sa = sinf(ang);
  float kf = (float)(2.0 * PI_D / 343.0) * fhz;
  float* out = V + (size_t)idx * 2 * N_MICS;
  #pragma unroll
  for (int m = 0; m < N_MICS; m++) {
    float ph = kf * (ca * pos[m*3+0] + sa * pos[m*3+1]);
    out[2*m]   = cosf(ph);
    out[2*m+1] = sinf(ph);
  }
}

// ---------------------------------------------------------------------------
// Kernel 2: fused per-(f,w): covariance -> Cholesky -> 36 MVDR solves ->
// WMMA beamform of this window's t-group(s) -> per-f partial power.
// One wave32 per block (EXEC all-ones required by WMMA).
// ---------------------------------------------------------------------------
__global__ void __launch_bounds__(32) mvdr_kernel(
    const float* __restrict__ xr_g, const float* __restrict__ xi_g,
    const float* __restrict__ V, float* __restrict__ partial)
{
  __shared__ float Xr[NSNAP * N_MICS], Xi[NSNAP * N_MICS];
  __shared__ cplx  Rm[64], Lm[64];
  __shared__ float As[16 * 48];   // stacked [Wr;Wi], rows k=0..15, cols a padded to 48

  const int lane = threadIdx.x;
  const int w = blockIdx.x, f = blockIdx.y;

  // Load window snapshots t = 8w .. 8w+31 (one snapshot per lane)
  {
    int t = w * STEP8 + lane;
    const float* pr = xr_g + ((size_t)t * F_FULL + (I0 + f)) * N_MICS;
    const float* pi = xi_g + ((size_t)t * F_FULL + (I0 + f)) * N_MICS;
    #pragma unroll
    for (int m = 0; m < 8; m++) { Xr[lane*8+m] = pr[m]; Xi[lane*8+m] = pi[m]; }
  }
  for (int i = lane; i < 16*48; i += 32) As[i] = 0.0f;
  __syncthreads();

  // Covariance R[i][j] = (1/32) sum_n x[n,i]*conj(x[n,j]) + diag loading
  for (int e = lane; e < 64; e += 32) {
    int i = e >> 3, j = e & 7;
    float sr = 0.f, si = 0.f;
    #pragma unroll
    for (int n = 0; n < NSNAP; n++) {
      float ari = Xr[n*8+i], aii = Xi[n*8+i];
      float arj = Xr[n*8+j], aij = Xi[n*8+j];
      sr += ari*arj + aii*aij;
      si += aii*arj - ari*aij;
    }
    sr *= (1.0f/NSNAP); si *= (1.0f/NSNAP);
    if (i == j) { sr += DIAG_LOAD; si = 0.f; }
    Rm[e] = {sr, si};
  }
  __syncthreads();

  // Complex Cholesky R = L L^H (lane 0; ~500 flops, negligible)
  if (lane == 0) {
    for (int j = 0; j < 8; j++) {
      float s = Rm[j*8+j].x;
      for (int k = 0; k < j; k++) { cplx l = Lm[j*8+k]; s -= l.x*l.x + l.y*l.y; }
      float d = sqrtf(fmaxf(s, 1e-20f));
      Lm[j*8+j] = {d, 0.f};
      float inv = 1.0f / d;
      for (int i2 = j+1; i2 < 8; i2++) {
        cplx acc = Rm[i2*8+j];
        for (int k = 0; k < j; k++) acc = csub(acc, cmul(Lm[i2*8+k], cconj(Lm[j*8+k])));
        Lm[i2*8+j] = cscale(acc, inv);
      }
    }
  }
  __syncthreads();

  // MVDR weights per angle: z = R^-1 v ; w = z / (v^H z)
  for (int a = lane; a < N_ANG; a += 32) {
    cplx v[8], z[8];
    const float* vp = V + (((size_t)f * N_ANG + a) * N_MICS) * 2;
    #pragma unroll
    for (int m = 0; m < 8; m++) v[m] = {vp[2*m], vp[2*m+1]};
    #pragma unroll
    for (int i = 0; i < 8; i++) {                 // forward: L y = v
      cplx acc = v[i];
      for (int k = 0; k < i; k++) acc = csub(acc, cmul(Lm[i*8+k], z[k]));
      z[i] = cscale(acc, 1.0f / Lm[i*8+i].x);
    }
    #pragma unroll
    for (int i = 7; i >= 0; i--) {                // backward: L^H z = y
      cplx acc = z[i];
      for (int k = i+1; k < 8; k++) acc = csub(acc, cmul(cconj(Lm[k*8+i]), z[k]));
      z[i] = cscale(acc, 1.0f / Lm[i*8+i].x);
    }
    cplx den = {0.f, 0.f};
    #pragma unroll
    for (int m = 0; m < 8; m++) den = cadd(den, cmul(cconj(v[m]), z[m]));
    float inv = 1.0f / (den.x*den.x + den.y*den.y);
    #pragma unroll
    for (int m = 0; m < 8; m++) {
      cplx wm = cscale(cmul(z[m], cconj(den)), inv);
      As[m*48 + a]     = wm.x;   // Wr rows 0..7
      As[(m+8)*48 + a] = wm.y;   // Wi rows 8..15
    }
  }
  __syncthreads();

  // Beamform: Yr = [Wr;Wi]^T [Xr;Xi], Yi = [Wr;Wi]^T [Xi;-Xr] via f32 WMMA
  const int nG = (w == 0) ? 4 : 1;
  const int t  = lane & 15;
  const bool hiH = (lane >= 16);
  const float mask = (t < 8) ? 1.0f : 0.0f;
  const int tc = (t < 8) ? t : 7;

  for (int gi = 0; gi < nG; gi++) {
    int rowbase = (w == 0) ? gi * 8 : 24;
    int tglob0  = (w == 0) ? gi * 8 : (w + 3) * 8;
    int rw = rowbase + tc;

    // B fragments for 4 K-steps (B VGPR v, lane L: row = 4k4 + v + (L>=16 ? 2:0), col = L&15)
    v2f Br[4], Bi[4];
    #pragma unroll
    for (int k4 = 0; k4 < 4; k4++) {
      int m0 = (k4*4 + (hiH ? 2 : 0)) & 7;
      int m1 = m0 + 1;
      if (k4 < 2) {  // rows 0..7 -> Xr / Xi
        Br[k4].x =  Xr[rw*8+m0]*mask;  Br[k4].y =  Xr[rw*8+m1]*mask;
        Bi[k4].x =  Xi[rw*8+m0]*mask;  Bi[k4].y =  Xi[rw*8+m1]*mask;
      } else {       // rows 8..15 -> Xi / -Xr
        Br[k4].x =  Xi[rw*8+m0]*mask;  Br[k4].y =  Xi[rw*8+m1]*mask;
        Bi[k4].x = -Xr[rw*8+m0]*mask;  Bi[k4].y = -Xr[rw*8+m1]*mask;
      }
    }

    for (int mt = 0; mt < 3; mt++) {      // angle tiles: 0..15, 16..31, 32..47(pad)
      int col = mt*16 + t;                // A lane M = L&15
      v8f accR = {}; v8f accI = {};
      #pragma unroll
      for (int k4 = 0; k4 < 4; k4++) {
        int k0 = k4*4 + (hiH ? 2 : 0);    // A VGPR v: K = 4k4 + v + (L>=16 ? 2:0)
        v2f Af;
        Af.x = As[k0*48 + col];
        Af.y = As[(k0+1)*48 + col];
        accR = __builtin_amdgcn_wmma_f32_16x16x4_f32(
                 false, Af, false, Br[k4], (short)0, accR, false, false);
        accI = __builtin_amdgcn_wmma_f32_16x16x4_f32(
                 false, Af, false, Bi[k4], (short)0, accI, false, false);
      }
      // D layout: VGPR r -> M = r (lanes 0-15) / r+8 (lanes 16-31), N = lane&15
      #pragma unroll
      for (int r = 0; r < 8; r++) {
        int aIdx = mt*16 + r + (hiH ? 8 : 0);
        if (aIdx < N_ANG && t < 8) {
          float yr = accR[r], yi = accI[r];
          partial[((size_t)f * T_FRAMES + (tglob0 + t)) * N_ANG + aIdx] = yr*yr + yi*yi;
        }
      }
    }
  }
}

// ---------------------------------------------------------------------------
// Kernel 3: power[t][a] = sum_f partial[f][t][a]
// ---------------------------------------------------------------------------
__global__ void reduce_kernel(const float* __restrict__ partial, float* __restrict__ out) {
  int idx = blockIdx.x * blockDim.x + threadIdx.x;
  if (idx >= T_FRAMES * N_ANG) return;
  float s = 0.0f;
  for (int f = 0; f < NF; f++) s += partial[(size_t)f * (T_FRAMES * N_ANG) + idx];
  out[idx] = s;
}

extern "C" void kernel_launch(void* const* d_in, const int* in_sizes, int n_in,
                              void* d_out, int out_size, void* d_ws, size_t ws_size,
                              hipStream_t stream) {
  const float* xr  = (const float*)d_in[0];
  const float* xi  = (const float*)d_in[1];
  const float* pos = (const float*)d_in[2];
  float* V       = (float*)d_ws;                                  // 428*36*8*2 floats (~1 MB)
  float* partial = V + (size_t)NF * N_ANG * N_MICS * 2;           // 428*512*36 floats (~31.6 MB)

  steer_kernel<<<(NF*N_ANG + 255)/256, 256, 0, stream>>>(pos, V);
  mvdr_kernel<<<dim3(NW, NF), 32, 0, stream>>>(xr, xi, V, partial);
  reduce_kernel<<<(T_FRAMES*N_ANG + 255)/256, 256, 0, stream>>>(partial, (float*)d_out);
}